// EdgeConv_41257455845941
// MI455X (gfx1250) — compile-verified
//
#include <hip/hip_runtime.h>

typedef __attribute__((ext_vector_type(2))) float v2f;
typedef __attribute__((ext_vector_type(8))) float v8f;

constexpr int   B_    = 4;
constexpr int   C_IN  = 32;
constexpr int   C_OUT = 64;
constexpr int   N_    = 32768;
constexpr int   KNN   = 16;
constexpr float EPS   = 1e-5f;

// -------- Pass 0: fold BN params once: bnp[m] = {inv1, sh1, inv2, sh2} -----
__global__ __launch_bounds__(64) void pass0_bn(
    const float* __restrict__ g,  const float* __restrict__ bta,
    const float* __restrict__ mu, const float* __restrict__ var,
    float4* __restrict__ bnp)
{
    const int m = threadIdx.x;            // 0..63
    const float inv1 = g[m]         / sqrtf(var[m]         + EPS);
    const float sh1  = bta[m]         - mu[m]         * inv1;
    const float inv2 = g[C_OUT + m] / sqrtf(var[C_OUT + m] + EPS);
    const float sh2  = bta[C_OUT + m] - mu[C_OUT + m] * inv2;
    bnp[m] = make_float4(inv1, sh1, inv2, sh2);
}

// -------- Pass 1: dual GEMM (W1,W2) via chained V_WMMA_F32_16X16X4_F32 ------
// One wave computes a 16(M) x 16(N) tile of BOTH local and edge.
// Epilogue: out[0:64] = relu(local*inv1+sh1); ws_edge = edge*inv2 (b,n,c);
//           ws_a = local*inv2 - sh2 (b,n,c).
__global__ __launch_bounds__(256) void pass1_gemm_bn(
    const float* __restrict__ feat, const float* __restrict__ W1,
    const float* __restrict__ W2,   const float4* __restrict__ bnp,
    float* __restrict__ out,
    float* __restrict__ ws_edge,    float* __restrict__ ws_a)
{
    const int w    = blockIdx.x * 8 + (threadIdx.x >> 5);
    const int lane = threadIdx.x & 31;
    const int half = lane >> 4;            // lane half selects K pair / M+8
    const int l16  = lane & 15;
    const int mt   = w & 3;                // 4 M-tiles of 16 cover C_OUT=64
    const int nt   = (w >> 2) & (N_/16 - 1);
    const int b    = w >> 13;              // 2 + log2(2048)
    const int nb   = nt * 16;
    const int mrow = mt * 16 + l16;        // A-matrix row held by this lane

    v8f c1 = {0.f,0.f,0.f,0.f,0.f,0.f,0.f,0.f};
    v8f c2 = c1;
    const float* fb = feat + (size_t)b * C_IN * N_;

    #pragma unroll
    for (int k0 = 0; k0 < C_IN; k0 += 4) {
        const int ka = k0 + 2 * half;      // lanes 0-15: K{k0,k0+1}; 16-31: K{k0+2,k0+3}
        v2f a1; a1.x = W1[mrow * C_IN + ka]; a1.y = W1[mrow * C_IN + ka + 1];
        v2f a2; a2.x = W2[mrow * C_IN + ka]; a2.y = W2[mrow * C_IN + ka + 1];
        v2f bv; bv.x = fb[(size_t)ka       * N_ + nb + l16];
                bv.y = fb[(size_t)(ka + 1) * N_ + nb + l16];
        // 8 args: (neg_a, A, neg_b, B, c_mod, C, reuse_a, reuse_b)
        c1 = __builtin_amdgcn_wmma_f32_16x16x4_f32(false, a1, false, bv, (short)0, c1, false, false);
        c2 = __builtin_amdgcn_wmma_f32_16x16x4_f32(false, a2, false, bv, (short)0, c2, false, false);
    }

    const int n = nb + l16;
    #pragma unroll
    for (int v = 0; v < 8; ++v) {
        const int m = mt * 16 + half * 8 + v;           // C/D layout: row = v + 8*half
        const float4 p = bnp[m];                        // {inv1, sh1, inv2, sh2}
        const float loc = c1[v], edg = c2[v];
        out[((size_t)b * 2 * C_OUT + m) * N_ + n] = fmaxf(loc * p.x + p.y, 0.f);
        ws_edge[((size_t)b * N_ + n) * C_OUT + m] = edg * p.z;
        ws_a  [((size_t)b * N_ + n) * C_OUT + m]  = loc * p.z - p.w;
    }
}

// -------- Pass 2: L2-resident gather + relu-mean over K -------------------
// One wave per point n: lane <-> channel pair => each gathered index reads a
// fully coalesced 256B row.  knn indices are wave-uniform (scalar loads).
// LDS transpose (stride 65, conflict-free) makes the final stores coalesced.
__global__ __launch_bounds__(256) void pass2_gather_mean(
    const long long* __restrict__ knn, const float* __restrict__ ws_edge,
    const float* __restrict__ ws_a,    float* __restrict__ out)
{
    __shared__ float tile[C_OUT * 65];
    const int blocksPerB = N_ / 64;
    const int b     = blockIdx.x / blocksPerB;
    const int nbase = (blockIdx.x % blocksPerB) * 64;
    // readfirstlane makes the wave id provably uniform -> n is uniform ->
    // knn loads become scalar (s_load, KMcnt), off the vector-memory path.
    const int wave  = __builtin_amdgcn_readfirstlane(threadIdx.x >> 5);
    const int lane  = threadIdx.x & 31;
    const int c0    = 2 * lane;

    #pragma unroll
    for (int i = 0; i < 8; ++i) {
        const int nl = wave * 8 + i;
        const int n  = nbase + nl;
        const float2 av = *(const float2*)(ws_a + ((size_t)b * N_ + n) * C_OUT + c0);
        float sx = 0.f, sy = 0.f;
        const long long* kp = knn + ((size_t)b * N_ + n) * KNN;
        #pragma unroll
        for (int k = 0; k < KNN; ++k) {
            const int idx = (int)kp[k];   // uniform address -> scalar load
            const float2 ev = *(const float2*)(ws_edge + ((size_t)b * N_ + idx) * C_OUT + c0);
            sx += fmaxf(ev.x - av.x, 0.f);
            sy += fmaxf(ev.y - av.y, 0.f);
        }
        tile[c0 * 65 + nl]       = sx;
        tile[(c0 + 1) * 65 + nl] = sy;
    }
    __syncthreads();
    const int t = threadIdx.x;
    #pragma unroll
    for (int r = 0; r < 16; ++r) {
        const int flat = r * 256 + t;
        const int c  = flat >> 6;
        const int nl = flat & 63;
        out[((size_t)b * 2 * C_OUT + C_OUT + c) * N_ + nbase + nl] =
            tile[c * 65 + nl] * (1.0f / 16.0f);
    }
}

extern "C" void kernel_launch(void* const* d_in, const int* in_sizes, int n_in,
                              void* d_out, int out_size, void* d_ws, size_t ws_size,
                              hipStream_t stream) {
    const float*     feat = (const float*)d_in[0];
    const long long* knn  = (const long long*)d_in[1];
    const float*     W1   = (const float*)d_in[2];
    const float*     W2   = (const float*)d_in[3];
    const float*     g    = (const float*)d_in[4];
    const float*     bta  = (const float*)d_in[5];
    const float*     mu   = (const float*)d_in[6];
    const float*     var  = (const float*)d_in[7];
    float* out     = (float*)d_out;
    float* ws_edge = (float*)d_ws;                       // B*N*64 floats (33.5 MB)
    float* ws_a    = ws_edge + (size_t)B_ * N_ * C_OUT;  // B*N*64 floats (33.5 MB)
    float4* bnp    = (float4*)(ws_a + (size_t)B_ * N_ * C_OUT); // 64 float4 (1 KB)

    pass0_bn<<<dim3(1), dim3(64), 0, stream>>>(g, bta, mu, var, bnp);

    const int waves1 = B_ * (N_ / 16) * 4;               // 32768 waves
    pass1_gemm_bn<<<dim3(waves1 / 8), dim3(256), 0, stream>>>(
        feat, W1, W2, bnp, out, ws_edge, ws_a);
    pass2_gather_mean<<<dim3(B_ * (N_ / 64)), dim3(256), 0, stream>>>(
        knn, ws_edge, ws_a, out);
}